// KVCache_15857019257359
// MI455X (gfx1250) — compile-verified
//
#include <hip/hip_runtime.h>

#define DEV __device__ __forceinline__

// ---------------- problem constants (from reference setup_inputs) ----------------
constexpr int B = 4, H = 32, P = 4096, D = 128;
constexpr int BH = B * H;             // 128
constexpr int RES = 128;              // RESIDUAL_LEN
constexpr int GS = 128;               // GROUP_SIZE
constexpr int TQ = P - RES;           // 3968 positions quantized
constexpr int G = TQ / GS;            // 31 seq groups
constexpr int RESROWS = 2 * RES + 1;  // 257

// ---------------- output layout ----------------
// Harness concatenates the returned tuple; int8 + bf16 promote to bf16, so every
// output element is stored as one bf16 (packed int8 codes are sign-extended ints).
constexpr size_t N_RES = (size_t)BH * RESROWS * D;   // 4,210,688
constexpr size_t N_Q   = (size_t)BH * TQ * (D / 2);  // 32,505,856
constexpr size_t N_SC  = (size_t)BH * G * D;         // 507,904 (== BH*TQ for v-scale)

constexpr size_t OFF_KRES = 0;
constexpr size_t OFF_VRES = OFF_KRES + N_RES;
constexpr size_t OFF_QKU  = OFF_VRES + N_RES;
constexpr size_t OFF_QKL  = OFF_QKU + N_Q;
constexpr size_t OFF_KSC  = OFF_QKL + N_Q;
constexpr size_t OFF_KMN  = OFF_KSC + N_SC;
constexpr size_t OFF_QVU  = OFF_KMN + N_SC;
constexpr size_t OFF_QVL  = OFF_QVU + N_Q;
constexpr size_t OFF_VSC  = OFF_QVL + N_Q;
constexpr size_t OFF_VMN  = OFF_VSC + N_SC;

// ---------------- bf16 helpers (bit-level, no header dependence) ----------------
DEV float bf2f(unsigned short u) {
    union { float f; unsigned i; } c; c.i = (unsigned)u << 16; return c.f;
}
DEV unsigned short f2bf(float f) {  // round-to-nearest-even
    union { float f; unsigned i; } c; c.f = f;
    unsigned i = c.i;
    return (unsigned short)((i + 0x7FFFu + ((i >> 16) & 1u)) >> 16);
}
DEV int qcode(float x, float mn, float inv) {
    float q = rintf((x - mn) * inv);          // jnp.round = ties-to-even
    q = fminf(fmaxf(q, 0.f), 255.f);
    return (int)q;
}
DEV unsigned short sbyte_bf(int b) {          // packed int8 code -> bf16 value
    return f2bf((float)(signed char)(b & 0xFF));
}

// ---- CDNA5 async global<->LDS (ASYNCcnt-tracked, 16B granule) ----
DEV void async_b128_to_lds(unsigned lds_byte_addr, unsigned long long gaddr) {
    asm volatile("global_load_async_to_lds_b128 %0, %1, off"
                 :: "v"(lds_byte_addr), "v"(gaddr) : "memory");
}
DEV void async_b128_from_lds(unsigned long long gaddr, unsigned lds_byte_addr) {
    asm volatile("global_store_async_from_lds_b128 %0, %1, off"
                 :: "v"(gaddr), "v"(lds_byte_addr) : "memory");
}
DEV void wait_async0() { asm volatile("s_wait_asynccnt 0" ::: "memory"); }

// ---- CDNA5 Tensor Data Mover: one-shot DMA of a contiguous 32KB tile to LDS ----
#if defined(__has_builtin)
#if __has_builtin(__builtin_amdgcn_tensor_load_to_lds)
#define HAVE_TDM 1
#endif
#endif

#ifdef HAVE_TDM
typedef unsigned int v4u __attribute__((ext_vector_type(4)));
typedef int          v4i __attribute__((ext_vector_type(4)));
typedef int          v8i __attribute__((ext_vector_type(8)));

DEV void tdm_load_tile_32k(unsigned lds_addr, unsigned long long ga) {
    // D# group 0 (CDNA5 ISA s8.3): count=1, lds_addr, global_addr, type=2
    v4u g0;
    g0.x = 1u;                                            // count=1, user descriptor
    g0.y = lds_addr;                                      // LDS byte address
    g0.z = (unsigned)(ga & 0xFFFFFFFFull);                // global_addr[31:0]
    g0.w = (unsigned)((ga >> 32) & 0x01FFFFFFull) | (2u << 30); // addr[56:32] | type=2
    // D# group 1 (s8.4): 1-D tile of 16384 x 2-byte elements (whole 32KB tile)
    v8i g1;
    g1[0] = 0x00010000;          // workgroup_mask=0 (not in cluster), data_size=1 (2B)
    g1[1] = (int)0x40000000;     // tensor_dim0[15:0]=0x4000 in bits[63:48]
    g1[2] = 0x00010000;          // tensor_dim0[31:16]=0, tensor_dim1[15:0]=1
    g1[3] = (int)0x40000000;     // tensor_dim1 hi=0, tile_dim0=0x4000 (bits[127:112])
    g1[4] = 1;                   // tile_dim1=1, tile_dim2=0
    g1[5] = 16384;               // tensor_dim0_stride[31:0]
    g1[6] = 0;                   // stride hi + tensor_dim1_stride lo
    g1[7] = 0;
    v4i z4 = {0, 0, 0, 0};
#if __clang_major__ >= 23
    v8i z8 = {0, 0, 0, 0, 0, 0, 0, 0};
    __builtin_amdgcn_tensor_load_to_lds(g0, g1, z4, z4, z8, 0);  // 6-arg (therock-10.0)
#else
    __builtin_amdgcn_tensor_load_to_lds(g0, g1, z4, z4, 0);      // 5-arg (ROCm 7.2)
#endif
}
#endif

// ======================= residual tail copy (async DMA through LDS) =======================
__global__ __launch_bounds__(256) void resid_copy(
    const uint4* __restrict__ k_new, const uint4* __restrict__ v_new,
    const uint4* __restrict__ k_buf, const uint4* __restrict__ v_buf,
    uint4* __restrict__ out)
{
    __shared__ alignas(16) unsigned char stage[256 * 16];  // one 16B slot per lane
    constexpr int CPT = BH * RESROWS * 16;     // 16B chunks per tensor (526,336 = 2056*256)
    int c = blockIdx.x * 256 + threadIdx.x;    // grid covers exactly 2*CPT
    const int which = (c >= CPT);              // uniform per block (CPT % 256 == 0)
    if (which) c -= CPT;
    const int wj = c & 15;                     // chunk within 128-elem row
    const int r  = c >> 4;                     // bh*257 + s
    const int s  = r % RESROWS;
    const int bh = r / RESROWS;

    const uint4* srcp;
    if (s < RES) srcp = &(which ? v_new : k_new)[((size_t)bh * P + TQ + s) * 16 + wj];
    else         srcp = &(which ? v_buf : k_buf)[(size_t)r * 16 + wj];
    uint4* dstp = &out[(which ? OFF_VRES : OFF_KRES) / 8 + (size_t)r * 16 + wj];

    const unsigned lds = (unsigned)(size_t)&stage[0] + (unsigned)threadIdx.x * 16u;
    async_b128_to_lds(lds, (unsigned long long)(size_t)srcp);   // global -> LDS
    wait_async0();                                              // LDS data landed
    async_b128_from_lds((unsigned long long)(size_t)dstp, lds); // LDS -> global
    // S_ENDPGM performs an implicit wait-idle, draining the async stores.
}

// ======================= K: quantize along seq, pack along d =======================
__global__ __launch_bounds__(256) void kquant(const unsigned short* __restrict__ k_new,
                                              unsigned short* __restrict__ out)
{
    __shared__ alignas(16) unsigned short tile[GS * D];   // 32KB row-major [l][d]
    __shared__ float pmn[2][D], pmx[2][D];
    __shared__ float s_mn[D], s_inv[D];

    const int t  = threadIdx.x;
    const int bh = blockIdx.x / G;
    const int g  = blockIdx.x % G;
    const unsigned short* src = k_new + ((size_t)bh * P + (size_t)g * GS) * D;
    const unsigned lds_base = (unsigned)(size_t)&tile[0];  // AS(3) offset = low 32 bits

#ifdef HAVE_TDM
    // One TDM descriptor moves the whole contiguous 32KB tile; only wave 0 issues it.
    if (t < 32) {
        tdm_load_tile_32k(lds_base, (unsigned long long)(size_t)src);
        __builtin_amdgcn_s_wait_tensorcnt(0);
    }
    __syncthreads();
#else
    const unsigned long long gbase = (unsigned long long)(size_t)src;
    #pragma unroll
    for (int i = 0; i < 8; ++i) {
        const int chunk = t + 256 * i;
        async_b128_to_lds(lds_base + (unsigned)chunk * 16u,
                          gbase + (unsigned long long)chunk * 16ull);
    }
    wait_async0();
    __syncthreads();
#endif

    // per-d min/max over the 128-seq group (lanes span consecutive d: conflict-free)
    {
        const int d = t & (D - 1), half = t >> 7;
        float mn = 3.0e38f, mx = -3.0e38f;
        const int l0 = half * 64;
        for (int l = l0; l < l0 + 64; ++l) {
            const float x = bf2f(tile[l * D + d]);
            mn = fminf(mn, x); mx = fmaxf(mx, x);
        }
        pmn[half][d] = mn; pmx[half][d] = mx;
    }
    __syncthreads();
    if (t < D) {
        const float mn = fminf(pmn[0][t], pmn[1][t]);
        const float mx = fmaxf(pmx[0][t], pmx[1][t]);
        const float sc = (mx - mn) * (1.0f / 255.0f);
        const float safe = (sc == 0.f) ? 1.f : sc;
        s_mn[t] = mn; s_inv[t] = 1.f / safe;
        out[OFF_KSC + ((size_t)bh * G + g) * D + t] = f2bf(sc);
        out[OFF_KMN + ((size_t)bh * G + g) * D + t] = f2bf(mn);
    }
    __syncthreads();

    // quantize + nibble-pack adjacent d; 4 packed codes (as bf16) per 8B store
    #pragma unroll
    for (int i = 0; i < 8; ++i) {
        const int idx = t + 256 * i;   // 0..2047
        const int l   = idx >> 4;
        const int wj  = idx & 15;      // d = wj*8 .. wj*8+7
        const uint4 raw = *(const uint4*)&tile[l * D + wj * 8];
        const unsigned w32[4] = {raw.x, raw.y, raw.z, raw.w};
        unsigned uw[2], lw[2];
        #pragma unroll
        for (int j = 0; j < 2; ++j) {
            unsigned short uh[2], lh[2];
            #pragma unroll
            for (int p = 0; p < 2; ++p) {
                const int d0 = wj * 8 + j * 4 + p * 2;
                const unsigned pr = w32[j * 2 + p];
                const int q0 = qcode(bf2f((unsigned short)(pr & 0xFFFF)), s_mn[d0],     s_inv[d0]);
                const int q1 = qcode(bf2f((unsigned short)(pr >> 16)),    s_mn[d0 + 1], s_inv[d0 + 1]);
                uh[p] = sbyte_bf(((q0 >> 4) << 4) | (q1 >> 4));
                lh[p] = sbyte_bf(((q0 & 15) << 4) | (q1 & 15));
            }
            uw[j] = (unsigned)uh[0] | ((unsigned)uh[1] << 16);
            lw[j] = (unsigned)lh[0] | ((unsigned)lh[1] << 16);
        }
        const size_t base = ((size_t)bh * TQ + (size_t)g * GS + l) * (D / 2) + wj * 4;
        *(uint2*)&out[OFF_QKU + base] = make_uint2(uw[0], uw[1]);
        *(uint2*)&out[OFF_QKL + base] = make_uint2(lw[0], lw[1]);
    }
}

// ======================= V: quantize along d, pack along seq (transposed out) =======================
constexpr int VSTR = 130;   // padded LDS stride: bank ~ (d + l/2) mod 64
__global__ __launch_bounds__(256) void vquant(const unsigned short* __restrict__ v_new,
                                              unsigned short* __restrict__ out)
{
    __shared__ unsigned short vt[D * VSTR];      // transposed tile [d][l], 33,280B
    __shared__ float pmn[GS][16], pmx[GS][16];   // partials per (l, d-chunk)
    __shared__ float s_mn[GS], s_inv[GS];

    const int t  = threadIdx.x;
    const int bh = blockIdx.x / G;
    const int g  = blockIdx.x % G;
    const unsigned short* src = v_new + ((size_t)bh * P + (size_t)g * GS) * D;

    const int dc = t & 15;    // 8-d chunk
    const int lg = t >> 4;    // base l
    #pragma unroll
    for (int i = 0; i < 8; ++i) {
        const int l = lg + 16 * i;
        const uint4 raw = *(const uint4*)&src[(size_t)l * D + dc * 8];  // coalesced b128
        const unsigned w[4] = {raw.x, raw.y, raw.z, raw.w};
        float mn = 3.0e38f, mx = -3.0e38f;
        #pragma unroll
        for (int j = 0; j < 4; ++j) {
            const unsigned short a = (unsigned short)(w[j] & 0xFFFF);
            const unsigned short b = (unsigned short)(w[j] >> 16);
            const float fa = bf2f(a), fb = bf2f(b);
            mn = fminf(mn, fminf(fa, fb)); mx = fmaxf(mx, fmaxf(fa, fb));
            const int d = dc * 8 + j * 2;
            vt[d * VSTR + l]       = a;          // element-level transpose into LDS
            vt[(d + 1) * VSTR + l] = b;
        }
        pmn[l][dc] = mn; pmx[l][dc] = mx;
    }
    __syncthreads();
    if (t < GS) {
        float mn = 3.0e38f, mx = -3.0e38f;
        #pragma unroll
        for (int c = 0; c < 16; ++c) { mn = fminf(mn, pmn[t][c]); mx = fmaxf(mx, pmx[t][c]); }
        const float sc = (mx - mn) * (1.0f / 255.0f);
        const float safe = (sc == 0.f) ? 1.f : sc;
        s_mn[t] = mn; s_inv[t] = 1.f / safe;
        out[OFF_VSC + (size_t)bh * TQ + (size_t)g * GS + t] = f2bf(sc);
        out[OFF_VMN + (size_t)bh * TQ + (size_t)g * GS + t] = f2bf(mn);
    }
    __syncthreads();

    // pack adjacent seq pairs per d row; output is [B,H,D, TQ/2]
    #pragma unroll
    for (int i = 0; i < 8; ++i) {
        const int idx = t + 256 * i;  // 0..2047
        const int d   = idx >> 4;
        const int wj  = idx & 15;     // l = wj*8 .. +7
        const int l0  = wj * 8;
        unsigned uw[2], lw[2];
        #pragma unroll
        for (int j = 0; j < 2; ++j) {
            unsigned short uh[2], lh[2];
            #pragma unroll
            for (int p = 0; p < 2; ++p) {
                const int l = l0 + j * 4 + p * 2;
                const int q0 = qcode(bf2f(vt[d * VSTR + l]),     s_mn[l],     s_inv[l]);
                const int q1 = qcode(bf2f(vt[d * VSTR + l + 1]), s_mn[l + 1], s_inv[l + 1]);
                uh[p] = sbyte_bf(((q0 >> 4) << 4) | (q1 >> 4));
                lh[p] = sbyte_bf(((q0 & 15) << 4) | (q1 & 15));
            }
            uw[j] = (unsigned)uh[0] | ((unsigned)uh[1] << 16);
            lw[j] = (unsigned)lh[0] | ((unsigned)lh[1] << 16);
        }
        const size_t base = ((size_t)bh * D + d) * (TQ / 2) + (size_t)g * (GS / 2) + wj * 4;
        *(uint2*)&out[OFF_QVU + base] = make_uint2(uw[0], uw[1]);
        *(uint2*)&out[OFF_QVL + base] = make_uint2(lw[0], lw[1]);
    }
}

// ======================= launcher =======================
extern "C" void kernel_launch(void* const* d_in, const int* in_sizes, int n_in,
                              void* d_out, int out_size, void* d_ws, size_t ws_size,
                              hipStream_t stream)
{
    (void)in_sizes; (void)n_in; (void)out_size; (void)d_ws; (void)ws_size;
    const unsigned short* k_new = (const unsigned short*)d_in[0];
    const unsigned short* v_new = (const unsigned short*)d_in[1];
    const unsigned short* k_buf = (const unsigned short*)d_in[2];
    const unsigned short* v_buf = (const unsigned short*)d_in[3];
    unsigned short* out = (unsigned short*)d_out;

    constexpr int RES_BLOCKS = (2 * BH * RESROWS * 16) / 256;  // 4112, exact
    resid_copy<<<RES_BLOCKS, 256, 0, stream>>>(
        (const uint4*)k_new, (const uint4*)v_new,
        (const uint4*)k_buf, (const uint4*)v_buf, (uint4*)out);
    kquant<<<BH * G, 256, 0, stream>>>(k_new, out);
    vquant<<<BH * G, 256, 0, stream>>>(v_new, out);
}